// Upsampling_56796647522704
// MI455X (gfx1250) — compile-verified
//
#include <hip/hip_runtime.h>

// ---------------------------------------------------------------------------
// CDNA5 (gfx1250) implementation. fp32 WMMA (V_WMMA_F32_16X16X4_F32) for all
// GEMMs; wave32 layouts per CDNA5 ISA 7.12.2. B-operand LDS tiles are stored
// k-pair-interleaved so every WMMA fragment is a single aligned ds_load_b64.
// Tile staging uses GLOBAL_LOAD_ASYNC_TO_LDS_B32 (ASYNCcnt) when available.
// ---------------------------------------------------------------------------

typedef float v2f __attribute__((ext_vector_type(2)));
typedef float v8f __attribute__((ext_vector_type(8)));

#define BB 256
#define DD 1024
#define PP 128
#define HH 5
#define LN_EPS 1e-5f

// B-side interleaved tiles: [k/2][NCOL][2]; NCOL=144 => k-row stride = 288
// dwords === 32 (mod 64 banks): the two 16-lane halves of a b64 fragment load
// touch disjoint bank ranges (conflict-free).
#define NCOL 144
// A-side tiles padded to an even stride so {A[m][k],A[m][k+1]} is aligned b64.
#define APAD 34

#if defined(__has_builtin)
#if __has_builtin(__builtin_amdgcn_global_load_async_to_lds_b32) && \
    __has_builtin(__builtin_amdgcn_s_wait_asynccnt)
#define USE_ASYNC_LDS 1
#endif
#endif
#ifndef USE_ASYNC_LDS
#define USE_ASYNC_LDS 0
#endif

typedef __attribute__((address_space(1))) int* gptr_i32;
typedef __attribute__((address_space(3))) int* lptr_i32;

__device__ __forceinline__ void stage_b32(const float* g, float* l) {
#if USE_ASYNC_LDS
  // Builtin signature (probed via diagnostic): (AS1 int*, AS3 int*, imm, imm).
  __builtin_amdgcn_global_load_async_to_lds_b32((gptr_i32)g, (lptr_i32)l, 0,
                                                0);
#else
  *l = *g;
#endif
}

__device__ __forceinline__ void stage_wait() {
#if USE_ASYNC_LDS
  __builtin_amdgcn_s_wait_asynccnt(0);  // this wave's LDS fills have landed
#endif
  __syncthreads();                      // all waves' fills visible
}

__device__ __forceinline__ v2f ld2(const float* p) {
  return *(const v2f*)p;  // 8B-aligned by construction
}

__device__ __forceinline__ v8f wmma4(v2f a, v2f b, v8f c) {
  // D = A(16x4,f32) * B(4x16,f32) + C(16x16,f32)
  return __builtin_amdgcn_wmma_f32_16x16x4_f32(
      /*neg_a=*/false, a, /*neg_b=*/false, b,
      /*c_mod=*/(short)0, c, /*reuse_a=*/false, /*reuse_b=*/false);
}

// ---------------------------------------------------------------------------
// Kernel 0: transpose pe_emb [P,D] -> peT [D,P] so kernel 1 reads coalesced.
// ---------------------------------------------------------------------------
__global__ __launch_bounds__(256) void k0_pe_t(const float* __restrict__ pe,
                                               float* __restrict__ peT) {
  int idx = blockIdx.x * 256 + threadIdx.x;  // grid covers D*P exactly
  int d = idx >> 7, p = idx & 127;
  peT[idx] = pe[(size_t)p * DD + d];
}

// ---------------------------------------------------------------------------
// Kernel 1: conv (K=4 patch GEMM) + pos-emb + exact GELU + LayerNorm1.
// One wave per (b,d) row of 128 values; writes conv (pre-GELU) and n1.
// ---------------------------------------------------------------------------
__global__ __launch_bounds__(256) void k1_front(
    const float* __restrict__ ts, const float* __restrict__ cw,
    const float* __restrict__ cb, const float* __restrict__ peT,
    const float* __restrict__ w1, const float* __restrict__ b1,
    float* __restrict__ conv, float* __restrict__ n1) {
  const int tid = threadIdx.x;
  const int wv = tid >> 5, lane = tid & 31;
  const int gw = blockIdx.x * 8 + wv;  // B*D waves total
  const int b = gw >> 10, d = gw & (DD - 1);

  const float4 kk = reinterpret_cast<const float4*>(cw)[d];  // conv_w[d,0,0:4]
  const float bias = cb[d];

  float cv[4], av[4];
#pragma unroll
  for (int j = 0; j < 4; ++j) {
    const int p = lane + 32 * j;
    const float4 x = reinterpret_cast<const float4*>(ts)[(size_t)b * PP + p];
    float v = x.x * kk.x + x.y * kk.y + x.z * kk.z + x.w * kk.w + bias +
              peT[(size_t)d * PP + p];
    cv[j] = v;                                                   // pre-GELU
    av[j] = 0.5f * v * (1.0f + erff(v * 0.70710678118654752f));  // exact GELU
  }
  float s = av[0] + av[1] + av[2] + av[3];
  float sq = av[0] * av[0] + av[1] * av[1] + av[2] * av[2] + av[3] * av[3];
#pragma unroll
  for (int m = 16; m >= 1; m >>= 1) {
    s += __shfl_xor(s, m, 32);
    sq += __shfl_xor(sq, m, 32);
  }
  const float mu = s * (1.0f / PP);
  const float var = sq * (1.0f / PP) - mu * mu;
  const float rstd = rsqrtf(var + LN_EPS);
  const size_t base = ((size_t)b * DD + d) * PP;
#pragma unroll
  for (int j = 0; j < 4; ++j) {
    const int p = lane + 32 * j;
    conv[base + p] = cv[j];
    n1[base + p] = (av[j] - mu) * rstd * w1[p] + b1[p];
  }
}

// ---------------------------------------------------------------------------
// Kernel 2: dense[b] = M[D,D] @ n1[b][D,P] + ts_emb[te[b]] (bcast over P).
// Wave w owns rows [16w,16w+16) x all 128 cols -> 8 fp32 WMMA accumulators.
// ---------------------------------------------------------------------------
__global__ __launch_bounds__(256) void k2_dense(
    const float* __restrict__ M, const float* __restrict__ n1,
    const float* __restrict__ ts_emb, const int* __restrict__ te,
    float* __restrict__ dense) {
  __shared__ float Ms[128][APAD];      // A tile, row-major, even stride
  __shared__ float Ns[16][NCOL][2];    // B tile, k-pair interleaved

  const int tid = threadIdx.x;
  const int wv = tid >> 5, lane = tid & 31;
  const int l15 = lane & 15, hl = lane >> 4;
  const int b = blockIdx.y;
  const int i0 = blockIdx.x * 128;

  v8f acc[8];
#pragma unroll
  for (int t = 0; t < 8; ++t)
#pragma unroll
    for (int r = 0; r < 8; ++r) acc[t][r] = 0.0f;

  for (int j0 = 0; j0 < DD; j0 += 32) {
    __syncthreads();
    for (int idx = tid; idx < 128 * 32; idx += 256) {
      int r = idx >> 5, c = idx & 31;
      stage_b32(&M[(size_t)(i0 + r) * DD + j0 + c], &Ms[r][c]);
    }
    for (int idx = tid; idx < 32 * 128; idx += 256) {
      int r = idx >> 7, c = idx & 127;
      stage_b32(&n1[((size_t)b * DD + j0 + r) * PP + c], &Ns[r >> 1][c][r & 1]);
    }
    stage_wait();
    const int rowA = 16 * wv + l15;
    for (int kk = 0; kk < 32; kk += 4) {
      const v2f a = ld2(&Ms[rowA][kk + 2 * hl]);
      const int k2 = (kk >> 1) + hl;
#pragma unroll
      for (int t = 0; t < 8; ++t)
        acc[t] = wmma4(a, ld2(&Ns[k2][16 * t + l15][0]), acc[t]);
    }
  }
  const int tb = te[b];
#pragma unroll
  for (int r = 0; r < 8; ++r) {
    const int row = i0 + 16 * wv + r + 8 * hl;
    const float tv = ts_emb[(size_t)tb * DD + row];
#pragma unroll
    for (int t = 0; t < 8; ++t)
      dense[((size_t)b * DD + row) * PP + 16 * t + l15] = acc[t][r] + tv;
  }
}

// ---------------------------------------------------------------------------
// Kernel 3: fused  x_ = W_h @ dense_b  and  corr = x_^T x_  per (b,h).
// x_ is produced in 64-row chunks into LDS (never hits HBM); the 128x128 corr
// accumulator lives in VGPRs (wave w holds corr rows [16w,16w+16) x 128).
// Epilogue: scale D^-1/2, causal mask (p>q -> -1e30), store TRANSPOSED
// corrT[b,h,q,p] so softmax over p is contiguous.
// ---------------------------------------------------------------------------
__global__ __launch_bounds__(256) void k3_corr(
    const float* __restrict__ attW, const float* __restrict__ dense,
    float* __restrict__ corrT) {
  __shared__ float Ws[64][APAD];       // GEMM1 A tile (att_W chunk)
  __shared__ float Ds[16][NCOL][2];    // GEMM1 B tile (dense), interleaved
  __shared__ float xcs[32][NCOL][2];   // x_ chunk (64 rows), interleaved:
                                       // serves as A and B of GEMM2

  const int tid = threadIdx.x;
  const int wv = tid >> 5, lane = tid & 31;
  const int l15 = lane & 15, hl = lane >> 4;
  const int h = blockIdx.x, b = blockIdx.y;
  const int rw = wv & 3, cw = wv >> 2;  // GEMM1 tile assignment (4x2 waves)

  v8f corrAcc[8];
#pragma unroll
  for (int t = 0; t < 8; ++t)
#pragma unroll
    for (int r = 0; r < 8; ++r) corrAcc[t][r] = 0.0f;

  for (int d0 = 0; d0 < DD; d0 += 64) {
    // ---- GEMM1: xc[64][128] = att_W[h][d0:d0+64][:] @ dense[b] ----
    v8f xcAcc[4];
#pragma unroll
    for (int t = 0; t < 4; ++t)
#pragma unroll
      for (int r = 0; r < 8; ++r) xcAcc[t][r] = 0.0f;

    for (int j0 = 0; j0 < DD; j0 += 32) {
      __syncthreads();
      for (int idx = tid; idx < 64 * 32; idx += 256) {
        int r = idx >> 5, c = idx & 31;
        stage_b32(&attW[((size_t)h * DD + d0 + r) * DD + j0 + c], &Ws[r][c]);
      }
      for (int idx = tid; idx < 32 * 128; idx += 256) {
        int r = idx >> 7, c = idx & 127;
        stage_b32(&dense[((size_t)b * DD + j0 + r) * PP + c],
                  &Ds[r >> 1][c][r & 1]);
      }
      stage_wait();
      const int rowA = 16 * rw + l15;
      for (int kk = 0; kk < 32; kk += 4) {
        const v2f a = ld2(&Ws[rowA][kk + 2 * hl]);
        const int k2 = (kk >> 1) + hl;
#pragma unroll
        for (int t = 0; t < 4; ++t)
          xcAcc[t] =
              wmma4(a, ld2(&Ds[k2][64 * cw + 16 * t + l15][0]), xcAcc[t]);
      }
    }
    __syncthreads();
#pragma unroll
    for (int r = 0; r < 8; ++r) {
      const int m = 16 * rw + r + 8 * hl;  // row of xc = K index of GEMM2
#pragma unroll
      for (int t = 0; t < 4; ++t)
        xcs[m >> 1][64 * cw + 16 * t + l15][m & 1] = xcAcc[t][r];
    }
    __syncthreads();

    // ---- GEMM2: corr += xc^T @ xc  (K = 64) ----
    for (int kk = 0; kk < 64; kk += 4) {
      const int k2 = (kk >> 1) + hl;
      const v2f a = ld2(&xcs[k2][16 * wv + l15][0]);
#pragma unroll
      for (int t = 0; t < 8; ++t)
        corrAcc[t] = wmma4(a, ld2(&xcs[k2][16 * t + l15][0]), corrAcc[t]);
    }
    __syncthreads();
  }

  const float scale = 0.03125f;  // 1024^(-1/2)
#pragma unroll
  for (int r = 0; r < 8; ++r) {
    const int p = 16 * wv + r + 8 * hl;
#pragma unroll
    for (int t = 0; t < 8; ++t) {
      const int q = 16 * t + l15;
      float v = corrAcc[t][r] * scale;
      if (p > q) v = -1e30f;  // causal mask, finite keeps softmax NaN-free
      corrT[(((size_t)b * HH + h) * PP + q) * PP + p] = v;
    }
  }
}

// ---------------------------------------------------------------------------
// Kernel 4: softmax over p (contiguous in corrT) and head collapse:
//   smw[b,p,q] = sum_h agg_w[h] * softmax_p(corrT[b,h,q,:])[p]
// One wave per (b,q).
// ---------------------------------------------------------------------------
__global__ __launch_bounds__(256) void k4_softmax(
    const float* __restrict__ corrT, const float* __restrict__ aggw,
    float* __restrict__ smw) {
  const int tid = threadIdx.x;
  const int wv = tid >> 5, lane = tid & 31;
  const int gw = blockIdx.x * 8 + wv;  // B*P waves
  const int b = gw >> 7, q = gw & 127;

  float acc[4] = {0.0f, 0.0f, 0.0f, 0.0f};
  for (int h = 0; h < HH; ++h) {
    const float* row = corrT + (((size_t)b * HH + h) * PP + q) * PP;
    float v[4];
#pragma unroll
    for (int j = 0; j < 4; ++j) v[j] = row[lane + 32 * j];
    float m = fmaxf(fmaxf(v[0], v[1]), fmaxf(v[2], v[3]));
#pragma unroll
    for (int k = 16; k >= 1; k >>= 1) m = fmaxf(m, __shfl_xor(m, k, 32));
    float e[4], s = 0.0f;
#pragma unroll
    for (int j = 0; j < 4; ++j) {
      e[j] = expf(v[j] - m);
      s += e[j];
    }
#pragma unroll
    for (int k = 16; k >= 1; k >>= 1) s += __shfl_xor(s, k, 32);
    const float g = aggw[h] / s;
#pragma unroll
    for (int j = 0; j < 4; ++j) acc[j] += e[j] * g;
  }
#pragma unroll
  for (int j = 0; j < 4; ++j) {
    const int p = lane + 32 * j;
    smw[((size_t)b * PP + p) * PP + q] = acc[j];
  }
}

// ---------------------------------------------------------------------------
// Kernel 5: agg[b] = dense[b] @ smw[b]  (K = 128), + agg_b + conv, then
// LayerNorm2 over the P axis fused in the epilogue (shfl reductions within
// 16-lane halves, matching the WMMA C layout).
// ---------------------------------------------------------------------------
__global__ __launch_bounds__(256) void k5_out(
    const float* __restrict__ dense, const float* __restrict__ smw,
    const float* __restrict__ conv, const float* __restrict__ w2,
    const float* __restrict__ b2, const float* __restrict__ aggb,
    float* __restrict__ out) {
  __shared__ float As[128][APAD];
  __shared__ float Bs[16][NCOL][2];

  const int tid = threadIdx.x;
  const int wv = tid >> 5, lane = tid & 31;
  const int l15 = lane & 15, hl = lane >> 4;
  const int b = blockIdx.y;
  const int i0 = blockIdx.x * 128;

  v8f acc[8];
#pragma unroll
  for (int t = 0; t < 8; ++t)
#pragma unroll
    for (int r = 0; r < 8; ++r) acc[t][r] = 0.0f;

  for (int j0 = 0; j0 < PP; j0 += 32) {
    __syncthreads();
    for (int idx = tid; idx < 128 * 32; idx += 256) {
      int r = idx >> 5, c = idx & 31;
      stage_b32(&dense[((size_t)b * DD + i0 + r) * PP + j0 + c], &As[r][c]);
    }
    for (int idx = tid; idx < 32 * 128; idx += 256) {
      int r = idx >> 7, c = idx & 127;
      stage_b32(&smw[((size_t)b * PP + j0 + r) * PP + c],
                &Bs[r >> 1][c][r & 1]);
    }
    stage_wait();
    const int rowA = 16 * wv + l15;
    for (int kk = 0; kk < 32; kk += 4) {
      const v2f a = ld2(&As[rowA][kk + 2 * hl]);
      const int k2 = (kk >> 1) + hl;
#pragma unroll
      for (int t = 0; t < 8; ++t)
        acc[t] = wmma4(a, ld2(&Bs[k2][16 * t + l15][0]), acc[t]);
    }
  }

  const float ab = aggb[0];
#pragma unroll
  for (int r = 0; r < 8; ++r) {
    const int row = i0 + 16 * wv + r + 8 * hl;
#pragma unroll
    for (int t = 0; t < 8; ++t)
      acc[t][r] += ab + conv[((size_t)b * DD + row) * PP + 16 * t + l15];
  }

  float w2v[8], b2v[8];
#pragma unroll
  for (int t = 0; t < 8; ++t) {
    w2v[t] = w2[16 * t + l15];
    b2v[t] = b2[16 * t + l15];
  }

#pragma unroll
  for (int r = 0; r < 8; ++r) {
    float s = 0.0f, sq = 0.0f;
#pragma unroll
    for (int t = 0; t < 8; ++t) {
      const float x = acc[t][r];
      s += x;
      sq += x * x;
    }
    // Row m's 128 values live in one 16-lane half x 8 tiles: reduce in-half.
#pragma unroll
    for (int k = 8; k >= 1; k >>= 1) {
      s += __shfl_xor(s, k, 32);
      sq += __shfl_xor(sq, k, 32);
    }
    const float mu = s * (1.0f / PP);
    const float var = sq * (1.0f / PP) - mu * mu;
    const float rstd = rsqrtf(var + LN_EPS);
    const int row = i0 + 16 * wv + r + 8 * hl;
#pragma unroll
    for (int t = 0; t < 8; ++t)
      out[((size_t)b * DD + row) * PP + 16 * t + l15] =
          (acc[t][r] - mu) * rstd * w2v[t] + b2v[t];
  }
}

// ---------------------------------------------------------------------------
extern "C" void kernel_launch(void* const* d_in, const int* in_sizes, int n_in,
                              void* d_out, int out_size, void* d_ws,
                              size_t ws_size, hipStream_t stream) {
  (void)in_sizes; (void)n_in; (void)out_size; (void)ws_size;

  const float* ts      = (const float*)d_in[0];
  const int*   te      = (const int*)d_in[1];
  const float* conv_w  = (const float*)d_in[2];
  const float* conv_b  = (const float*)d_in[3];
  const float* pe_emb  = (const float*)d_in[4];
  const float* ln1_w   = (const float*)d_in[5];
  const float* ln1_b   = (const float*)d_in[6];
  const float* dense_M = (const float*)d_in[7];
  const float* ts_emb  = (const float*)d_in[8];
  const float* ln2_w   = (const float*)d_in[9];
  const float* ln2_b   = (const float*)d_in[10];
  const float* att_W   = (const float*)d_in[11];
  const float* agg_w   = (const float*)d_in[12];
  const float* agg_b   = (const float*)d_in[13];
  float* out = (float*)d_out;

  // Workspace layout (floats). The n1 region is dead after kernel 2 and is
  // reused for corrT (84MB) + smw (17MB). Total = 3*B*D*P + D*P floats ~403MB.
  float* ws = (float*)d_ws;
  const size_t NBDP = (size_t)BB * DD * PP;  // 33,554,432
  float* conv    = ws;
  float* scratch = ws + NBDP;
  float* dense   = ws + 2 * NBDP;
  float* peT     = ws + 3 * NBDP;
  float* n1    = scratch;
  float* corrT = scratch;                    // B*H*P*P = 20,971,520
  float* smw   = scratch + (size_t)BB * HH * PP * PP;

  k0_pe_t<<<(DD * PP) / 256, 256, 0, stream>>>(pe_emb, peT);
  k1_front<<<(BB * DD) / 8, 256, 0, stream>>>(ts, conv_w, conv_b, peT, ln1_w,
                                              ln1_b, conv, n1);
  k2_dense<<<dim3(DD / 128, BB), 256, 0, stream>>>(dense_M, n1, ts_emb, te,
                                                   dense);
  k3_corr<<<dim3(HH, BB), 256, 0, stream>>>(att_W, dense, corrT);
  k4_softmax<<<(BB * PP) / 8, 256, 0, stream>>>(corrT, agg_w, smw);
  k5_out<<<dim3(DD / 128, BB), 256, 0, stream>>>(dense, smw, conv, ln2_w,
                                                 ln2_b, agg_b, out);
}